// GATRecommenderWithPrompt_9861244912153
// MI455X (gfx1250) — compile-verified
//
#include <hip/hip_runtime.h>
#include <hip/hip_bf16.h>

typedef float v2f __attribute__((ext_vector_type(2)));
typedef float v8f __attribute__((ext_vector_type(8)));

#define DDIM 64
#define GEMM_ROWS 128          // rows per block (8 waves x 16 rows)
#define SX_STRIDE 68           // 64 + 4 pad: conflict-free (4k mod 64 distinct), float4-aligned
#define SW_STRIDE 68

// ---- float ordering key for atomic max on uint (monotone wrt float compare) ----
__device__ __forceinline__ unsigned fkey(float f) {
    unsigned u = __float_as_uint(f);
    return (u & 0x80000000u) ? ~u : (u | 0x80000000u);
}
__device__ __forceinline__ float fdec(unsigned k) {
    return (k & 0x80000000u) ? __uint_as_float(k & 0x7FFFFFFFu)
                             : __uint_as_float(~k);
}
#define FKEY_NEG_INF 0x007FFFFFu   // fkey(-inf)

// native f32 atomic add (global_atomic_add_f32) — avoid CAS-loop fallback
__device__ __forceinline__ void fadd_atomic(float* p, float v) {
    unsafeAtomicAdd(p, v);
}

// ---------------- prompt projection: p[d] = sum_j prompt[j]*proj_w[d,j] + proj_b[d] ----------------
__global__ void compute_pvec(const float* __restrict__ prompt, const float* __restrict__ pw,
                             const float* __restrict__ pb, float* __restrict__ pvec, int P) {
    int d = threadIdx.x;
    if (d >= DDIM) return;
    float s = pb[d];
    for (int j = 0; j < P; ++j) s += prompt[j] * pw[d * P + j];
    pvec[d] = s;
}

// ---------------- x = emb + p (broadcast); also seed out_sum = x ----------------
__global__ __launch_bounds__(256) void add_prompt(const float* __restrict__ emb,
                                                  const float* __restrict__ pvec,
                                                  float* __restrict__ x,
                                                  float* __restrict__ out_sum,
                                                  long long total) {
    long long i = (long long)blockIdx.x * blockDim.x + threadIdx.x;
    if (i >= total) return;
    float v = emb[i] + pvec[i & (DDIM - 1)];
    x[i] = v;
    out_sum[i] = v;
}

// ---------------- xl = x @ W^T via V_WMMA_F32_16X16X4_F32 ----------------
// A (16x4 f32, 2 VGPRs): lane L<16: M=L, K={k0,k0+1}; lane>=16: M=L-16, K={k0+2,k0+3}
// B (4x16, 2 VGPRs): same lane->K mapping, lane%16 = N column
// D (16x16, 8 VGPRs): reg v, lanes 0-15 -> M=v, lanes 16-31 -> M=v+8; N=lane%16
// k-outer loop: A loaded once per k-step, 4 independent accumulators (no same-D
// back-to-back RAW chain; halves LDS fragment traffic vs c-outer).
__global__ __launch_bounds__(256) void gemm_xw_t(const float* __restrict__ x,
                                                 const float* __restrict__ W,   // [64 x 64] row-major (out,k)
                                                 float* __restrict__ xl, int N) {
    __shared__ float sx[GEMM_ROWS * SX_STRIDE];   // ~34.8 KB
    __shared__ float sw[DDIM * SW_STRIDE];        // ~17.4 KB
    const int t = threadIdx.x;
    const int rowBase = blockIdx.x * GEMM_ROWS;

    // stage W (coalesced float4)
    {
        const float4* Wv = (const float4*)W;
        float4* swv = (float4*)sw;
        for (int i = t; i < (DDIM * DDIM) / 4; i += 256) {
            int r = i >> 4, c4 = i & 15;
            swv[r * (SW_STRIDE / 4) + c4] = Wv[i];
        }
    }
    // stage x tile (coalesced float4; clamp OOB rows)
    {
        const float4* xv = (const float4*)x;
        float4* sxv = (float4*)sx;
        for (int i = t; i < (GEMM_ROWS * DDIM) / 4; i += 256) {
            int r = i >> 4, c4 = i & 15;
            int gr = rowBase + r;
            if (gr >= N) gr = N - 1;
            sxv[r * (SX_STRIDE / 4) + c4] = xv[(long long)gr * 16 + c4];
        }
    }
    __syncthreads();

    const int wave  = t >> 5;            // 0..7
    const int lane  = t & 31;
    const int lrow  = lane & 15;         // M (A) / N (B) index
    const int khalf = (lane >> 4) << 1;  // 0 or 2
    const int wrow  = wave * 16;         // row offset of this wave's tile

    v8f acc0 = {0.f,0.f,0.f,0.f,0.f,0.f,0.f,0.f};
    v8f acc1 = acc0, acc2 = acc0, acc3 = acc0;

    const float* arow = &sx[(wrow + lrow) * SX_STRIDE + khalf];
    const float* brow = &sw[lrow * SW_STRIDE + khalf];

    #pragma unroll
    for (int k0 = 0; k0 < DDIM; k0 += 4) {
        v2f a, b0, b1, b2, b3;
        a.x  = arow[k0];                       a.y  = arow[k0 + 1];
        b0.x = brow[k0];                       b0.y = brow[k0 + 1];
        b1.x = brow[16 * SW_STRIDE + k0];      b1.y = brow[16 * SW_STRIDE + k0 + 1];
        b2.x = brow[32 * SW_STRIDE + k0];      b2.y = brow[32 * SW_STRIDE + k0 + 1];
        b3.x = brow[48 * SW_STRIDE + k0];      b3.y = brow[48 * SW_STRIDE + k0 + 1];
        acc0 = __builtin_amdgcn_wmma_f32_16x16x4_f32(false, a, false, b0, (short)0, acc0, false, false);
        acc1 = __builtin_amdgcn_wmma_f32_16x16x4_f32(false, a, false, b1, (short)0, acc1, false, false);
        acc2 = __builtin_amdgcn_wmma_f32_16x16x4_f32(false, a, false, b2, (short)0, acc2, false, false);
        acc3 = __builtin_amdgcn_wmma_f32_16x16x4_f32(false, a, false, b3, (short)0, acc3, false, false);
    }

    // epilogue: fast path (full tile, uniform branch, EXEC stays all-1s)
    const int rhi = (lane >> 4) * 8;
    const long long rbase = (long long)(rowBase + wrow + rhi);
    if (rowBase + GEMM_ROWS <= N) {
        float* p = xl + rbase * DDIM + lrow;
        #pragma unroll
        for (int v = 0; v < 8; ++v) {
            p[(long long)v * DDIM +  0] = acc0[v];
            p[(long long)v * DDIM + 16] = acc1[v];
            p[(long long)v * DDIM + 32] = acc2[v];
            p[(long long)v * DDIM + 48] = acc3[v];
        }
    } else {
        #pragma unroll
        for (int v = 0; v < 8; ++v) {
            long long gr = rbase + v;
            if (gr < N) {
                float* p = xl + gr * DDIM + lrow;
                p[ 0] = acc0[v];
                p[16] = acc1[v];
                p[32] = acc2[v];
                p[48] = acc3[v];
            }
        }
    }
}

// ---------------- a_src[n], a_dst[n]: warp-per-node dot products ----------------
__global__ __launch_bounds__(256) void att_dots(const float* __restrict__ xl,
                                                const float* __restrict__ av_s,
                                                const float* __restrict__ av_d,
                                                float* __restrict__ a_src,
                                                float* __restrict__ a_dst, int N) {
    long long tid = (long long)blockIdx.x * blockDim.x + threadIdx.x;
    int n = (int)(tid >> 5);
    int lane = threadIdx.x & 31;
    if (n >= N) return;
    float2 v = ((const float2*)(xl + (long long)n * DDIM))[lane];
    float2 s = ((const float2*)av_s)[lane];
    float2 d = ((const float2*)av_d)[lane];
    float ps = v.x * s.x + v.y * s.y;
    float pd = v.x * d.x + v.y * d.y;
    #pragma unroll
    for (int off = 16; off > 0; off >>= 1) {
        ps += __shfl_xor(ps, off, 32);
        pd += __shfl_xor(pd, off, 32);
    }
    if (lane == 0) { a_src[n] = ps; a_dst[n] = pd; }
}

// ---------------- per-layer scratch init (acc aliases x buffer; x is dead here) ----------------
__global__ __launch_bounds__(256) void init_layer(float* __restrict__ acc,
                                                  unsigned* __restrict__ segmax,
                                                  float* __restrict__ segsum, int N) {
    int n = blockIdx.x * blockDim.x + threadIdx.x;
    if (n >= N) return;
    segmax[n] = FKEY_NEG_INF;
    segsum[n] = 0.f;
    float4 z = {0.f, 0.f, 0.f, 0.f};
    float4* p = (float4*)(acc + (long long)n * DDIM);
    #pragma unroll
    for (int i = 0; i < DDIM / 4; ++i) p[i] = z;
}

__device__ __forceinline__ void edge_pair(const int* __restrict__ ei, long long e,
                                          int E, int& s, int& d) {
    if (e < E) { s = ei[e]; d = ei[(long long)E + e]; }
    else       { s = d = (int)(e - E); }
}

// ---------------- pass 1: segment max of leaky-relu logits ----------------
__global__ __launch_bounds__(256) void edge_max(const int* __restrict__ ei, int E, int N,
                                                const float* __restrict__ a_src,
                                                const float* __restrict__ a_dst,
                                                unsigned* __restrict__ segmax) {
    long long e = (long long)blockIdx.x * blockDim.x + threadIdx.x;
    long long tot = (long long)E + N;
    if (e >= tot) return;
    int s, d; edge_pair(ei, e, E, s, d);
    float v = a_src[s] + a_dst[d];
    v = v > 0.f ? v : 0.2f * v;
    atomicMax(&segmax[d], fkey(v));
}

// ---------------- pass 2: segment sum of exp(logit - max) ----------------
__global__ __launch_bounds__(256) void edge_sum(const int* __restrict__ ei, int E, int N,
                                                const float* __restrict__ a_src,
                                                const float* __restrict__ a_dst,
                                                const unsigned* __restrict__ segmax,
                                                float* __restrict__ segsum) {
    long long e = (long long)blockIdx.x * blockDim.x + threadIdx.x;
    long long tot = (long long)E + N;
    if (e >= tot) return;
    int s, d; edge_pair(ei, e, E, s, d);
    float v = a_src[s] + a_dst[d];
    v = v > 0.f ? v : 0.2f * v;
    float m = fdec(segmax[d]);
    fadd_atomic(&segsum[d], __expf(v - m));
}

// ---------------- pass 3: warp-per-edge weighted scatter: acc[dst] += alpha * xl[src] ----------------
__global__ __launch_bounds__(256) void edge_scatter(const int* __restrict__ ei, int E, int N,
                                                    const float* __restrict__ a_src,
                                                    const float* __restrict__ a_dst,
                                                    const unsigned* __restrict__ segmax,
                                                    const float* __restrict__ segsum,
                                                    const float* __restrict__ xl,
                                                    float* __restrict__ acc) {
    long long tid = (long long)blockIdx.x * blockDim.x + threadIdx.x;
    long long e = tid >> 5;
    int lane = threadIdx.x & 31;
    long long tot = (long long)E + N;
    if (e >= tot) return;
    int s, d; edge_pair(ei, e, E, s, d);
    float v = a_src[s] + a_dst[d];
    v = v > 0.f ? v : 0.2f * v;
    float m = fdec(segmax[d]);
    float alpha = __expf(v - m) / (segsum[d] + 1e-16f);
    float2 xv = ((const float2*)(xl + (long long)s * DDIM))[lane];
    float* dstp = acc + (long long)d * DDIM + 2 * lane;
    fadd_atomic(dstp,     xv.x * alpha);
    fadd_atomic(dstp + 1, xv.y * alpha);
}

// ---------------- x = elu(acc + bias); out_sum += x ----------------
__global__ __launch_bounds__(256) void finalize_layer(float* __restrict__ xbuf,
                                                      const float* __restrict__ bias,
                                                      float* __restrict__ out_sum,
                                                      long long total) {
    long long i = (long long)blockIdx.x * blockDim.x + threadIdx.x;
    if (i >= total) return;
    float v = xbuf[i] + bias[i & (DDIM - 1)];
    v = v > 0.f ? v : (__expf(v) - 1.f);
    xbuf[i] = v;
    out_sum[i] += v;
}

__global__ __launch_bounds__(256) void scale_out(float* __restrict__ out, float inv,
                                                 long long total) {
    long long i = (long long)blockIdx.x * blockDim.x + threadIdx.x;
    if (i >= total) return;
    out[i] *= inv;
}

static inline unsigned gridFor(long long work, int block) {
    return (unsigned)((work + block - 1) / block);
}

extern "C" void kernel_launch(void* const* d_in, const int* in_sizes, int n_in,
                              void* d_out, int out_size, void* d_ws, size_t ws_size,
                              hipStream_t stream) {
    const int*   ei       = (const int*)d_in[0];
    const float* emb      = (const float*)d_in[1];
    const float* prompt   = (const float*)d_in[2];
    const float* proj_w   = (const float*)d_in[3];
    const float* proj_b   = (const float*)d_in[4];
    const float* conv_ws  = (const float*)d_in[5];
    const float* att_srcs = (const float*)d_in[6];
    const float* att_dsts = (const float*)d_in[7];
    const float* conv_bs  = (const float*)d_in[8];

    const int E = in_sizes[0] / 2;
    const int N = in_sizes[1] / DDIM;
    const int P = in_sizes[2];
    const int L = in_sizes[5] / (DDIM * DDIM);

    float* out = (float*)d_out;

    // workspace carve-up
    float*    pvec   = (float*)d_ws;                       // 64
    float*    xbuf   = pvec + 64;                          // N*D (also acc)
    float*    xlb    = xbuf + (size_t)N * DDIM;            // N*D
    float*    a_src  = xlb + (size_t)N * DDIM;             // N
    float*    a_dst  = a_src + N;                          // N
    unsigned* segmax = (unsigned*)(a_dst + N);             // N
    float*    segsum = (float*)(segmax + N);               // N

    const long long total = (long long)N * DDIM;
    const long long totE  = (long long)E + N;

    compute_pvec<<<1, 64, 0, stream>>>(prompt, proj_w, proj_b, pvec, P);
    add_prompt<<<gridFor(total, 256), 256, 0, stream>>>(emb, pvec, xbuf, out, total);

    for (int l = 0; l < L; ++l) {
        const float* W  = conv_ws  + (size_t)l * DDIM * DDIM;
        const float* as = att_srcs + (size_t)l * DDIM;
        const float* ad = att_dsts + (size_t)l * DDIM;
        const float* cb = conv_bs  + (size_t)l * DDIM;

        gemm_xw_t<<<gridFor(N, GEMM_ROWS), 256, 0, stream>>>(xbuf, W, xlb, N);
        att_dots<<<gridFor((long long)N * 32, 256), 256, 0, stream>>>(xlb, as, ad, a_src, a_dst, N);
        init_layer<<<gridFor(N, 256), 256, 0, stream>>>(xbuf, segmax, segsum, N);
        edge_max<<<gridFor(totE, 256), 256, 0, stream>>>(ei, E, N, a_src, a_dst, segmax);
        edge_sum<<<gridFor(totE, 256), 256, 0, stream>>>(ei, E, N, a_src, a_dst, segmax, segsum);
        edge_scatter<<<gridFor(totE * 32, 256), 256, 0, stream>>>(ei, E, N, a_src, a_dst,
                                                                  segmax, segsum, xlb, xbuf);
        finalize_layer<<<gridFor(total, 256), 256, 0, stream>>>(xbuf, cb, out, total);
    }
    scale_out<<<gridFor(total, 256), 256, 0, stream>>>(out, 1.f / (float)(L + 1), total);
}